// GCN_Model_77077483095003
// MI455X (gfx1250) — compile-verified
//
#include <hip/hip_runtime.h>
#include <hip/hip_bf16.h>

typedef __attribute__((ext_vector_type(2))) float v2f;
typedef __attribute__((ext_vector_type(8))) float v8f;

#define D128 128
#define NGRAPH 256
#define NOUT 10
#define KCHUNK 32
#define LSTRIDE 144   // 32-row chunk of W, padded so half-wave bank sets are disjoint

// ---------------------------------------------------------------------------
// fill: p[i] = v
__global__ void k_fill(float* __restrict__ p, long n, float v) {
    long i = (long)blockIdx.x * blockDim.x + threadIdx.x;
    if (i < n) p[i] = v;
}

// deg accumulation over dst (deg buffer pre-filled with 1.0 for self-loops)
__global__ void k_degree(const int* __restrict__ dst, float* __restrict__ deg, int E) {
    int i = blockIdx.x * blockDim.x + threadIdx.x;
    if (i < E) atomicAdd(&deg[dst[i]], 1.0f);
}

// dinv = rsqrt(deg) in place (deg >= 1 always due to self-loop)
__global__ void k_rsqrt(float* __restrict__ d, int n) {
    int i = blockIdx.x * blockDim.x + threadIdx.x;
    if (i < n) d[i] = rsqrtf(d[i]);
}

// ---------------------------------------------------------------------------
// GEMM: O[N,128] = A[N,128] @ W[128,128], fp32 WMMA 16x16x4.
// One wave computes a 16-row x 128-col strip (8 accumulator tiles).
// W staged through LDS in 32-row chunks.
__global__ __launch_bounds__(256) void k_gemm(const float* __restrict__ A,
                                              const float* __restrict__ W,
                                              float* __restrict__ O, int nrows) {
    __shared__ float w_lds[KCHUNK * LSTRIDE];   // 32*144*4 = 18.4 KB

    const int wave = threadIdx.x >> 5;
    const int lane = threadIdx.x & 31;
    const int strip = blockIdx.x * 8 + wave;
    const bool valid = (strip * 16) < nrows;    // wave-uniform

    const int rlo  = lane & 15;   // row within tile for A, col within tile for B/D
    const int kp   = lane >> 4;   // half-wave selector
    const int row  = strip * 16 + rlo;

    v8f acc[8] = {};
    const float* arow = A + (size_t)row * D128;

    for (int k0 = 0; k0 < D128; k0 += KCHUNK) {
        __syncthreads();
        // cooperative stage of W chunk into LDS
        for (int i = threadIdx.x; i < KCHUNK * D128; i += 256) {
            int r = i >> 7, c = i & 127;
            w_lds[r * LSTRIDE + c] = W[(k0 + r) * D128 + c];
        }
        __syncthreads();

        if (valid) {
            #pragma unroll
            for (int k = 0; k < KCHUNK; k += 4) {
                const int ka = k + 2 * kp;   // A: lanes 0-15 hold K=k..k+1, lanes 16-31 K=k+2..k+3
                v2f a;
                a.x = arow[k0 + ka];
                a.y = arow[k0 + ka + 1];
                #pragma unroll
                for (int t = 0; t < 8; ++t) {
                    const int col = t * 16 + rlo;
                    v2f b;
                    b.x = w_lds[ka * LSTRIDE + col];
                    b.y = w_lds[(ka + 1) * LSTRIDE + col];
                    acc[t] = __builtin_amdgcn_wmma_f32_16x16x4_f32(
                        false, a, false, b, (short)0, acc[t], false, false);
                }
            }
        }
    }

    if (valid) {
        // D layout: vgpr r, lane: m = r + 8*(lane/16), n = lane%16
        const size_t base = (size_t)(strip * 16) * D128;
        #pragma unroll
        for (int t = 0; t < 8; ++t) {
            float* op = O + base + (size_t)(t * 16 + rlo);
            #pragma unroll
            for (int r = 0; r < 8; ++r)
                op[(size_t)(r + 8 * kp) * D128] = acc[t][r];
        }
    }
}

// ---------------------------------------------------------------------------
// Edge scatter: one wave per edge (edges [0,E) real, [E,E+N) self-loops).
// msg = T[src] * dinv[src]*dinv[dst]; atomic-add into acc[dst].
__global__ __launch_bounds__(256) void k_scatter(const float* __restrict__ T,
                                                 const int* __restrict__ src,
                                                 const int* __restrict__ dst,
                                                 const float* __restrict__ dinv,
                                                 float* __restrict__ acc,
                                                 int E, int N) {
    const int wid  = (int)(((long)blockIdx.x * blockDim.x + threadIdx.x) >> 5);
    const int lane = threadIdx.x & 31;
    if (wid >= E + N) return;
    int s, d;
    if (wid < E) { s = src[wid]; d = dst[wid]; }
    else         { s = d = wid - E; }
    const float nrm = dinv[s] * dinv[d];
    float4 v = ((const float4*)(T + (size_t)s * D128))[lane];
    v.x *= nrm; v.y *= nrm; v.z *= nrm; v.w *= nrm;
    float* ap = acc + (size_t)d * D128 + lane * 4;
    atomicAdd(ap + 0, v.x);
    atomicAdd(ap + 1, v.y);
    atomicAdd(ap + 2, v.z);
    atomicAdd(ap + 3, v.w);
}

// h = relu(h + bias[col]) in place
__global__ void k_bias_relu(float* __restrict__ h, const float* __restrict__ b, long n) {
    long i = (long)blockIdx.x * blockDim.x + threadIdx.x;
    if (i < n) {
        float v = h[i] + b[i & 127];
        h[i] = v > 0.0f ? v : 0.0f;
    }
}

// ---------------------------------------------------------------------------
// Pooling: one wave per node. Post-ReLU values >= 0, so u32 atomicMax on the
// float bit pattern (0-initialized) implements float segment-max exactly.
__global__ __launch_bounds__(256) void k_pool(const float* __restrict__ h,
                                              const int* __restrict__ batch,
                                              unsigned int* __restrict__ gmax,
                                              float* __restrict__ gsum,
                                              float* __restrict__ cnt, int N) {
    const int wid  = (int)(((long)blockIdx.x * blockDim.x + threadIdx.x) >> 5);
    const int lane = threadIdx.x & 31;
    if (wid >= N) return;
    const int g = batch[wid];
    float4 v = ((const float4*)(h + (size_t)wid * D128))[lane];
    unsigned int* mp = gmax + (size_t)g * D128 + lane * 4;
    float*        sp = gsum + (size_t)g * D128 + lane * 4;
    atomicMax(mp + 0, __float_as_uint(v.x));
    atomicMax(mp + 1, __float_as_uint(v.y));
    atomicMax(mp + 2, __float_as_uint(v.z));
    atomicMax(mp + 3, __float_as_uint(v.w));
    atomicAdd(sp + 0, v.x);
    atomicAdd(sp + 1, v.y);
    atomicAdd(sp + 2, v.z);
    atomicAdd(sp + 3, v.w);
    if (lane == 0) atomicAdd(&cnt[g], 1.0f);
}

// out[b,o] = bias[o] + sum_d gmax*W[d] + (gsum/cnt)*W[128+d] + gsum*W[256+d]
__global__ void k_out(const float* __restrict__ gmax, const float* __restrict__ gsum,
                      const float* __restrict__ cnt, const float* __restrict__ W,
                      const float* __restrict__ bias, float* __restrict__ out) {
    int i = blockIdx.x * blockDim.x + threadIdx.x;
    if (i >= NGRAPH * NOUT) return;
    const int b = i / NOUT, o = i % NOUT;
    float c = cnt[b];
    c = c > 1.0f ? c : 1.0f;
    float a = bias[o];
    for (int d = 0; d < D128; ++d) {
        const float mx = gmax[b * D128 + d];
        const float s  = gsum[b * D128 + d];
        a += mx * W[d * NOUT + o] + (s / c) * W[(D128 + d) * NOUT + o]
           + s * W[(2 * D128 + d) * NOUT + o];
    }
    out[i] = a;
}

// ---------------------------------------------------------------------------
static inline int grid1(long n) { return (int)((n + 255) / 256); }

extern "C" void kernel_launch(void* const* d_in, const int* in_sizes, int n_in,
                              void* d_out, int out_size, void* d_ws, size_t ws_size,
                              hipStream_t stream) {
    const float* x      = (const float*)d_in[0];
    const int*   ei     = (const int*)d_in[1];   // [2, E]: row0 = src, row1 = dst
    const int*   batch  = (const int*)d_in[2];
    const float* conv_w = (const float*)d_in[3]; // [3,128,128]
    const float* conv_b = (const float*)d_in[4]; // [3,128]
    const float* out_w  = (const float*)d_in[5]; // [384,10]
    const float* out_b  = (const float*)d_in[6]; // [10]

    const int N = in_sizes[0] / D128;
    const int E = in_sizes[1] / 2;
    const int* src = ei;
    const int* dst = ei + E;

    // workspace layout
    float* t    = (float*)d_ws;                      // [N,128] gemm output
    float* acc  = t + (size_t)N * D128;              // [N,128] scatter accumulator
    float* dinv = acc + (size_t)N * D128;            // [N]
    float* gmax = dinv + N;                          // [256,128] (u32 max bits)
    float* gsum = gmax + NGRAPH * D128;              // [256,128]
    float* cnt  = gsum + NGRAPH * D128;              // [256]

    // symmetric normalization: deg (incl. self-loop) -> rsqrt
    k_fill<<<grid1(N), 256, 0, stream>>>(dinv, N, 1.0f);
    k_degree<<<grid1(E), 256, 0, stream>>>(dst, dinv, E);
    k_rsqrt<<<grid1(N), 256, 0, stream>>>(dinv, N);

    const int strips = (N + 15) / 16;                // N=100000 -> 6250, exact
    const float* cur = x;
    for (int l = 0; l < 3; ++l) {
        k_gemm<<<(strips + 7) / 8, 256, 0, stream>>>(cur, conv_w + (size_t)l * D128 * D128, t, N);
        k_fill<<<grid1((long)N * D128), 256, 0, stream>>>(acc, (long)N * D128, 0.0f);
        const long waves = (long)E + N;
        k_scatter<<<(int)((waves + 7) / 8), 256, 0, stream>>>(t, src, dst, dinv, acc, E, N);
        k_bias_relu<<<grid1((long)N * D128), 256, 0, stream>>>(acc, conv_b + (size_t)l * D128, (long)N * D128);
        cur = acc;
    }

    // pooling + head
    k_fill<<<grid1(NGRAPH * D128 * 2 + NGRAPH), 256, 0, stream>>>(gmax, NGRAPH * D128 * 2 + NGRAPH, 0.0f);
    k_pool<<<(int)(((long)N + 7) / 8), 256, 0, stream>>>(acc, batch, (unsigned int*)gmax, gsum, cnt, N);
    k_out<<<grid1(NGRAPH * NOUT), 256, 0, stream>>>(gmax, gsum, cnt, out_w, out_b, (float*)d_out);
}